// MultiCompartmentCapsuleLayer_5832565588491
// MI455X (gfx1250) — compile-verified
//
#include <hip/hip_runtime.h>
#include <hip/hip_bf16.h>
#include <math.h>

typedef float v2f __attribute__((ext_vector_type(2)));
typedef float v4f __attribute__((ext_vector_type(4)));
typedef float v8f __attribute__((ext_vector_type(8)));

#define RNODES 32768
#define NCAP   10
#define NB     16
#define NCB    160      // NCAP*NB
#define NBLK   64       // r-chunk blocks per (c,b)
#define RCHUNK 512      // RNODES / NBLK

// L2 residency split: priors is 336 MB, L2 is 192 MB. Keep cb < CB_RESIDENT
// (80 * 2 MB = 160 MB) in L2 with regular (RT) loads/stores; stream the rest
// with non-temporal (TH=NT) accesses so they don't evict the resident set.
// logits (20 MB, RT) also stays resident -> ~180 MB hot set < 192 MB.
#define CB_RESIDENT 80
#define C_RESIDENT  5    // cb = c*16+b < 80  <=>  c < 5

// ---------------------------------------------------------------------------
// zero-fill (logits init)
// ---------------------------------------------------------------------------
__global__ void zero_f32_kernel(float* __restrict__ p, int n4) {
    int i = blockIdx.x * blockDim.x + threadIdx.x;
    if (i < n4) ((float4*)p)[i] = make_float4(0.f, 0.f, 0.f, 0.f);
}

// ---------------------------------------------------------------------------
// priors[c*16+b][r][o] = sum_i x[b,r,i] * W[c,r,i,o]   via V_WMMA_F32_16X16X4_F32
// grid: (R/64, C), block: 256 (8 waves), each wave does 8 consecutive r values
// ---------------------------------------------------------------------------
__global__ void priors_wmma_kernel(const float* __restrict__ x,
                                   const float* __restrict__ w,
                                   float* __restrict__ priors) {
    const int c    = blockIdx.y;
    const int wave = threadIdx.x >> 5;
    const int lane = threadIdx.x & 31;
    const int half = lane >> 4;      // 0: lanes 0-15, 1: lanes 16-31
    const int l16  = lane & 15;

    const int rbase = (blockIdx.x * 8 + wave) * 8;   // 8 r per wave
    const float* wc = w + (size_t)c * RNODES * 128;  // W[c,:,:,:]
    const bool nt  = (c >= C_RESIDENT);              // stream-out portion

    for (int rr = 0; rr < 8; ++rr) {
        const int r = rbase + rr;

        // A: 16x8 (b x i). VGPR0 holds K=2*half, VGPR1 holds K=2*half+1.
        const float* xp = x + ((size_t)l16 * RNODES + r) * 8 + 2 * half;
        float2 xa = *(const float2*)(xp);       // i = 2h, 2h+1
        float2 xb = *(const float2*)(xp + 4);   // i = 4+2h, 4+2h+1
        v2f a0 = { xa.x, xa.y };
        v2f a1 = { xb.x, xb.y };

        // B: 8x16 (i x o), row-striped across lanes, same K pattern as A.
        // W is read exactly once -> non-temporal so it doesn't thrash L2.
        const float* wp = wc + (size_t)r * 128 + (2 * half) * 16 + l16;
        v2f b0 = { __builtin_nontemporal_load(wp),
                   __builtin_nontemporal_load(wp + 16) };   // rows 2h, 2h+1
        v2f b1 = { __builtin_nontemporal_load(wp + 64),
                   __builtin_nontemporal_load(wp + 80) };   // rows 4+2h, 5+2h

        v8f acc = {};
        acc = __builtin_amdgcn_wmma_f32_16x16x4_f32(false, a0, false, b0,
                                                    (short)0, acc, false, false);
        acc = __builtin_amdgcn_wmma_f32_16x16x4_f32(false, a1, false, b1,
                                                    (short)0, acc, false, false);

        // D: VGPR v -> (b = v + 8*half, o = l16)
        float* dp = priors + (((size_t)(c * 16 + 8 * half) * RNODES + r) * 16) + l16;
        if (nt) {
#pragma unroll
            for (int v = 0; v < 8; ++v)
                __builtin_nontemporal_store(acc[v], dp + (size_t)v * RNODES * 16);
        } else {
#pragma unroll
            for (int v = 0; v < 8; ++v)
                dp[(size_t)v * RNODES * 16] = acc[v];
        }
    }
}

// ---------------------------------------------------------------------------
// Routing pass A: fused delta-update + flash-softmax partials.
// grid: (NBLK, NCB), block: 256.
// Thread t handles o-group (t&3)*4..+3 and r = j*64 + t/4 for j in 0..7.
// ---------------------------------------------------------------------------
__global__ void routing_pass_a(const float* __restrict__ priors,
                               float* __restrict__ logits,
                               const float* __restrict__ outv,   // [NCB*16]
                               float* __restrict__ part_m,
                               float* __restrict__ part_z,
                               float* __restrict__ part_s,
                               int first_iter) {
    const int cb  = blockIdx.y;
    const int blk = blockIdx.x;
    const int tid = threadIdx.x;

    const float* p  = priors + ((size_t)cb * RNODES + (size_t)blk * RCHUNK) * 16;
    float*       lg = logits + (size_t)cb * RNODES + blk * RCHUNK;
    const bool   nt = (cb >= CB_RESIDENT);

    const int og = (tid & 3) * 4;
    float4 ov = make_float4(0.f, 0.f, 0.f, 0.f);
    if (!first_iter) ov = *(const float4*)(outv + cb * 16 + og);

    float m = -3.0e38f, Z = 0.f;
    float4 s = make_float4(0.f, 0.f, 0.f, 0.f);

#pragma unroll
    for (int j = 0; j < 8; ++j) {
        const int idx = j * 1024 + tid * 4;
        const int r   = idx >> 4;
        v4f pvv;
        if (nt) pvv = __builtin_nontemporal_load((const v4f*)(p + idx));
        else    pvv = *(const v4f*)(p + idx);
        float4 pv = make_float4(pvv.x, pvv.y, pvv.z, pvv.w);

        float l = 0.f;
        if (!first_iter) {
            // delta_r = sum_o priors[r,o]*out[o], partial over this thread's 4 o's
            float d = pv.x * ov.x + pv.y * ov.y + pv.z * ov.z + pv.w * ov.w;
            d += __shfl_xor(d, 1, 32);
            d += __shfl_xor(d, 2, 32);
            l = lg[r] + d;
            if ((tid & 3) == 0) lg[r] = l;   // one writer per r
        }

        // online softmax accumulation
        float mnew  = fmaxf(m, l);
        float scale = expf(m - mnew);
        float e     = expf(l - mnew);
        Z   = Z * scale + e;
        s.x = s.x * scale + e * pv.x;
        s.y = s.y * scale + e * pv.y;
        s.z = s.z * scale + e * pv.z;
        s.w = s.w * scale + e * pv.w;
        m = mnew;
    }

    // block-wide max of m
    __shared__ float red_m[256];
    red_m[tid] = m;
    __syncthreads();
    for (int off = 128; off > 0; off >>= 1) {
        if (tid < off) red_m[tid] = fmaxf(red_m[tid], red_m[tid + off]);
        __syncthreads();
    }
    const float M = red_m[0];
    __syncthreads();

    const float f = expf(m - M);

    __shared__ float sacc[16];
    __shared__ float zacc;
    if (tid < 16) sacc[tid] = 0.f;
    if (tid == 0) zacc = 0.f;
    __syncthreads();

    atomicAdd(&sacc[og + 0], s.x * f);
    atomicAdd(&sacc[og + 1], s.y * f);
    atomicAdd(&sacc[og + 2], s.z * f);
    atomicAdd(&sacc[og + 3], s.w * f);
    if ((tid & 3) == 0) atomicAdd(&zacc, Z * f);   // each r counted once
    __syncthreads();

    const int pidx = cb * NBLK + blk;
    if (tid < 16)  part_s[pidx * 16 + tid] = sacc[tid];
    if (tid == 16) part_z[pidx] = zacc;
    if (tid == 17) part_m[pidx] = M;
}

// ---------------------------------------------------------------------------
// Routing pass B: merge NBLK partials per (c,b), normalize, squash.
// grid: NCB, block: 64 (one thread per partial).
// ---------------------------------------------------------------------------
__global__ void routing_pass_b(const float* __restrict__ part_m,
                               const float* __restrict__ part_z,
                               const float* __restrict__ part_s,
                               float* __restrict__ outdst) {
    const int cb = blockIdx.x;
    const int t  = threadIdx.x;   // 0..63

    float m = part_m[cb * NBLK + t];

    __shared__ float sm[64];
    sm[t] = m;
    __syncthreads();
    for (int off = 32; off > 0; off >>= 1) {
        if (t < off) sm[t] = fmaxf(sm[t], sm[t + off]);
        __syncthreads();
    }
    const float M = sm[0];
    __syncthreads();

    const float f = expf(m - M);

    __shared__ float sz[64];
    sz[t] = part_z[cb * NBLK + t] * f;
    __syncthreads();
    for (int off = 32; off > 0; off >>= 1) {
        if (t < off) sz[t] += sz[t + off];
        __syncthreads();
    }
    const float Z = sz[0];
    __syncthreads();

    __shared__ float sacc[16];
    if (t < 16) sacc[t] = 0.f;
    __syncthreads();
#pragma unroll
    for (int o = 0; o < 16; ++o)
        atomicAdd(&sacc[o], part_s[(cb * NBLK + t) * 16 + o] * f);
    __syncthreads();

    const float S = (t < 16) ? sacc[t] / Z : 0.f;
    __shared__ float Svec[16];
    if (t < 16) Svec[t] = S;
    __syncthreads();
    if (t < 16) {
        float sq = 0.f;
#pragma unroll
        for (int o = 0; o < 16; ++o) sq += Svec[o] * Svec[o];
        const float scale = (sq / (1.f + sq)) / sqrtf(sq);
        outdst[cb * 16 + t] = scale * S;
    }
}

// ---------------------------------------------------------------------------
extern "C" void kernel_launch(void* const* d_in, const int* in_sizes, int n_in,
                              void* d_out, int out_size, void* d_ws, size_t ws_size,
                              hipStream_t stream) {
    const float* x = (const float*)d_in[0];          // [16, 32768, 8]
    const float* w = (const float*)d_in[1];          // [10, 32768, 8, 16]
    float* out = (float*)d_out;                      // [10, 16, 16] flat

    float* priors = (float*)d_ws;                              // NCB*R*16
    float* logits = priors + (size_t)NCB * RNODES * 16;        // NCB*R
    float* part_m = logits + (size_t)NCB * RNODES;             // NCB*NBLK
    float* part_z = part_m + (size_t)NCB * NBLK;               // NCB*NBLK
    float* part_s = part_z + (size_t)NCB * NBLK;               // NCB*NBLK*16
    float* outbuf = part_s + (size_t)NCB * NBLK * 16;          // NCB*16

    // logits = 0
    {
        const int n4 = (NCB * RNODES) / 4;
        zero_f32_kernel<<<(n4 + 255) / 256, 256, 0, stream>>>(logits, n4);
    }

    // priors GEMM (WMMA)
    priors_wmma_kernel<<<dim3(RNODES / 64, NCAP), 256, 0, stream>>>(x, w, priors);

    // dynamic routing: 10 iterations, one priors pass each
    for (int i = 0; i < NCAP; ++i) {
        routing_pass_a<<<dim3(NBLK, NCB), 256, 0, stream>>>(
            priors, logits, outbuf, part_m, part_z, part_s, (i == 0) ? 1 : 0);
        float* dst = (i == NCAP - 1) ? out : outbuf;
        routing_pass_b<<<dim3(NCB), 64, 0, stream>>>(part_m, part_z, part_s, dst);
    }
}